// PointNetSetAbstraction_317827580572
// MI455X (gfx1250) — compile-verified
//
#include <hip/hip_runtime.h>
#include <hip/hip_bf16.h>
#include <stdint.h>

// Problem constants
#define B_      8
#define N_      4096
#define NPOINT_ 1024
#define NSAMPLE_ 32
#define CFEAT_  64
#define CIN_    67
#define K0PAD   68
#define MTOT    (B_ * NSAMPLE_ * NPOINT_)   // 262144 columns
#define EPS_    1e-5f

typedef float v2f __attribute__((ext_vector_type(2)));
typedef float v4f __attribute__((ext_vector_type(4)));
typedef float v8f __attribute__((ext_vector_type(8)));

// Activation layout: channel-group-of-4 packed.  Element (c, m) lives at
//   X[((c>>2)*MTOT + m)*4 + (c&3)]
// -> a WMMA B-fragment lane load (k, k+1) is one contiguous 8B load, and the
//    D-tile rows a lane owns pack into 16B stores.

__device__ __forceinline__ unsigned long long ullmin(unsigned long long a, unsigned long long b) {
    return a < b ? a : b;
}

// ---------------------------------------------------------------------------
// new_xyz = gather(xyz, fps_idx)
// ---------------------------------------------------------------------------
__global__ void newxyz_kernel(const float* __restrict__ xyz, const int* __restrict__ fps,
                              float* __restrict__ out) {
    int p = blockIdx.x * blockDim.x + threadIdx.x;
    if (p < B_ * NPOINT_) {
        int b = p >> 10;
        int j = fps[p];
        const float* s = &xyz[((size_t)b * N_ + j) * 3];
        out[p * 3 + 0] = s[0];
        out[p * 3 + 1] = s[1];
        out[p * 3 + 2] = s[2];
    }
}

// ---------------------------------------------------------------------------
// zero-pad layer-0 weights (64 x 67) -> (64 x 68)
// ---------------------------------------------------------------------------
__global__ void padw0_kernel(const float* __restrict__ w0, float* __restrict__ wp0) {
    int t = blockIdx.x * blockDim.x + threadIdx.x;
    if (t < 64 * K0PAD) {
        int oc = t / K0PAD, c = t % K0PAD;
        wp0[t] = (c < CIN_) ? w0[oc * CIN_ + c] : 0.0f;
    }
}

// ---------------------------------------------------------------------------
// kNN: 32 smallest squared distances per centroid, stable ascending order.
// Key = (f32_bits(d2) << 32) | point_index  (d2 >= 0 -> bit-order monotonic,
// index in low bits reproduces argsort stable tie-break). One block / centroid.
// ---------------------------------------------------------------------------
__global__ __launch_bounds__(256) void knn_kernel(const float* __restrict__ xyz,
                                                  const int* __restrict__ fps,
                                                  int* __restrict__ knn) {
    __shared__ unsigned long long part[256];
    const int b  = blockIdx.x >> 10;
    const int np = blockIdx.x & 1023;
    const int t  = threadIdx.x;

    const int ci = fps[b * NPOINT_ + np];
    const float cx = xyz[((size_t)b * N_ + ci) * 3 + 0];
    const float cy = xyz[((size_t)b * N_ + ci) * 3 + 1];
    const float cz = xyz[((size_t)b * N_ + ci) * 3 + 2];

    unsigned long long key[16];
#pragma unroll
    for (int i = 0; i < 16; ++i) {
        int j = t + 256 * i;
        const float* p = &xyz[((size_t)b * N_ + j) * 3];
        float dx = p[0] - cx, dy = p[1] - cy, dz = p[2] - cz;
        float d2 = dx * dx + dy * dy + dz * dz;
        key[i] = ((unsigned long long)__float_as_uint(d2) << 32) | (unsigned int)j;
    }

    for (int r = 0; r < NSAMPLE_; ++r) {
        unsigned long long lm = key[0];
#pragma unroll
        for (int i = 1; i < 16; ++i) lm = ullmin(lm, key[i]);
        part[t] = lm;
        __syncthreads();
        for (int off = 128; off > 0; off >>= 1) {
            if (t < off) part[t] = ullmin(part[t], part[t + off]);
            __syncthreads();
        }
        unsigned long long gm = part[0];
        __syncthreads();
        if (t == 0) knn[(b * NPOINT_ + np) * NSAMPLE_ + r] = (int)(gm & 0xffffffffull);
#pragma unroll
        for (int i = 0; i < 16; ++i)
            if (key[i] == gm) key[i] = ~0ull;
    }
}

// ---------------------------------------------------------------------------
// Build x0 in packed layout, 17 channel-groups {xyz(3), points(64), pad(1)}.
// m = (b*32 + s)*1024 + np.  One b128 store per group, coalesced across m.
// ---------------------------------------------------------------------------
__global__ __launch_bounds__(256) void gather_kernel(const float* __restrict__ xyz,
                                                     const float* __restrict__ pts,
                                                     const int* __restrict__ knn,
                                                     float* __restrict__ x0) {
    int m = blockIdx.x * blockDim.x + threadIdx.x;
    if (m >= MTOT) return;
    int b  = m >> 15;
    int s  = (m >> 10) & 31;
    int np = m & 1023;
    int j  = knn[(b * NPOINT_ + np) * NSAMPLE_ + s];
    const float* px = &xyz[((size_t)b * N_ + j) * 3];
    const float* pf = &pts[((size_t)b * N_ + j) * CFEAT_];

    float ch[K0PAD];
    ch[0] = px[0];
    ch[1] = px[1];
    ch[2] = px[2];
#pragma unroll
    for (int c = 0; c < CFEAT_; ++c) ch[3 + c] = pf[c];
    ch[K0PAD - 1] = 0.0f;

#pragma unroll
    for (int kk = 0; kk < K0PAD / 4; ++kk) {
        v4f t = {ch[4 * kk + 0], ch[4 * kk + 1], ch[4 * kk + 2], ch[4 * kk + 3]};
        *(v4f*)(x0 + ((size_t)kk * MTOT + m) * 4) = t;
    }
}

// ---------------------------------------------------------------------------
// Conv (1x1) via V_WMMA_F32_16X16X4_F32 on packed activations.
// MODE 0: stats pass  (Sum y, Sum y^2 per channel; no store)
// MODE 1: normalize pass (relu(scale*y+shift) -> packed X_next, 2 b128/lane/s)
// MODE 2: normalize + max-over-s fused, store (B,NPOINT,OC), 2 b128/lane
// Grid: (np_tiles=64, OC/16, B), block = 1 wave (32 lanes).
// ---------------------------------------------------------------------------
template <int KPAD, int OC, int MODE>
__global__ __launch_bounds__(32) void conv_kernel(const float* __restrict__ X,
                                                  const float* __restrict__ W,
                                                  const float* __restrict__ bias,
                                                  const float* __restrict__ scale,
                                                  const float* __restrict__ shift,
                                                  float* __restrict__ sum,
                                                  float* __restrict__ sumsq,
                                                  float* __restrict__ out) {
    const int lane = threadIdx.x;
    const int half = lane >> 4;
    const int l16  = lane & 15;
    const int np0  = blockIdx.x * 16;
    const int oc0  = blockIdx.y * 16;
    const int b    = blockIdx.z;

    // A (weight) fragments: lane holds (oc = oc0 + l16, k = 4*kk + 2*half + v).
    // Row-major W, the (k, k+1) pair is contiguous & 8B aligned -> b64 load.
    v2f afrag[KPAD / 4];
#pragma unroll
    for (int kk = 0; kk < KPAD / 4; ++kk)
        afrag[kk] = *(const v2f*)(W + (size_t)(oc0 + l16) * KPAD + kk * 4 + 2 * half);

    float bv[8], scv[8], shv[8];
#pragma unroll
    for (int v = 0; v < 8; ++v) {
        int oc = oc0 + v + 8 * half;
        bv[v] = bias[oc];
        if (MODE != 0) { scv[v] = scale[oc]; shv[v] = shift[oc]; }
    }

    float s1[8], s2[8], mx[8];
#pragma unroll
    for (int v = 0; v < 8; ++v) { s1[v] = 0.0f; s2[v] = 0.0f; mx[v] = 0.0f; }

    // Output channel-group bases for the two 4-row packs this lane owns.
    const int gLo = (oc0 + 8 * half) >> 2;      // rows oc0+8h .. +3
    const int gHi = gLo + 1;                    // rows oc0+8h+4 .. +7

    for (int s = 0; s < NSAMPLE_; ++s) {
        const int m0 = ((b * NSAMPLE_ + s) << 10) + np0;
        const int mcol = m0 + l16;
        v8f c;
#pragma unroll
        for (int v = 0; v < 8; ++v) c[v] = bv[v];   // accumulate on top of bias
#pragma unroll
        for (int kk = 0; kk < KPAD / 4; ++kk) {
            // B fragment: (k, k+1) adjacent inside the channel group -> b64 load
            v2f bf = *(const v2f*)(X + ((size_t)kk * MTOT + mcol) * 4 + 2 * half);
            c = __builtin_amdgcn_wmma_f32_16x16x4_f32(false, afrag[kk], false, bf,
                                                      (short)0, c, false, false);
        }
        if (MODE == 0) {
#pragma unroll
            for (int v = 0; v < 8; ++v) { float y = c[v]; s1[v] += y; s2[v] += y * y; }
        } else if (MODE == 1) {
            v4f ylo = {fmaxf(c[0] * scv[0] + shv[0], 0.0f), fmaxf(c[1] * scv[1] + shv[1], 0.0f),
                       fmaxf(c[2] * scv[2] + shv[2], 0.0f), fmaxf(c[3] * scv[3] + shv[3], 0.0f)};
            v4f yhi = {fmaxf(c[4] * scv[4] + shv[4], 0.0f), fmaxf(c[5] * scv[5] + shv[5], 0.0f),
                       fmaxf(c[6] * scv[6] + shv[6], 0.0f), fmaxf(c[7] * scv[7] + shv[7], 0.0f)};
            *(v4f*)(out + ((size_t)gLo * MTOT + mcol) * 4) = ylo;
            *(v4f*)(out + ((size_t)gHi * MTOT + mcol) * 4) = yhi;
        } else {
#pragma unroll
            for (int v = 0; v < 8; ++v) {
                float y = fmaxf(c[v] * scv[v] + shv[v], 0.0f);
                mx[v] = fmaxf(mx[v], y);
            }
        }
    }

    if (MODE == 0) {
        // reduce the 16 columns (within each 16-lane half) -> per-channel partials
#pragma unroll
        for (int v = 0; v < 8; ++v) {
            float a = s1[v], q = s2[v];
#pragma unroll
            for (int msk = 1; msk < 16; msk <<= 1) {
                a += __shfl_xor(a, msk, 32);
                q += __shfl_xor(q, msk, 32);
            }
            if (l16 == 0) {
                atomicAdd(&sum[oc0 + v + 8 * half], a);
                atomicAdd(&sumsq[oc0 + v + 8 * half], q);
            }
        }
    } else if (MODE == 2) {
        // out is (B, NPOINT, OC): oc fastest -> two contiguous b128 per lane
        float* dst = out + (size_t)(b * NPOINT_ + np0 + l16) * OC + oc0 + 8 * half;
        v4f mlo = {mx[0], mx[1], mx[2], mx[3]};
        v4f mhi = {mx[4], mx[5], mx[6], mx[7]};
        *(v4f*)(dst) = mlo;
        *(v4f*)(dst + 4) = mhi;
    }
}

// ---------------------------------------------------------------------------
// BatchNorm finalize: scale = g/sqrt(var+eps), shift = beta - mean*scale
// ---------------------------------------------------------------------------
__global__ void finalize_kernel(const float* __restrict__ sum, const float* __restrict__ sumsq,
                                const float* __restrict__ g, const float* __restrict__ beta,
                                float* __restrict__ scale, float* __restrict__ shift, int oc_n) {
    int t = threadIdx.x;
    if (t < oc_n) {
        const float inv = 1.0f / (float)MTOT;
        float mean = sum[t] * inv;
        float var  = sumsq[t] * inv - mean * mean;
        float sc   = g[t] / sqrtf(var + EPS_);
        scale[t] = sc;
        shift[t] = beta[t] - mean * sc;
    }
}

__global__ void zero_stats_kernel(float* __restrict__ sum, float* __restrict__ sumsq) {
    int t = threadIdx.x;
    sum[t] = 0.0f;
    sumsq[t] = 0.0f;
}

// ---------------------------------------------------------------------------
extern "C" void kernel_launch(void* const* d_in, const int* in_sizes, int n_in,
                              void* d_out, int out_size, void* d_ws, size_t ws_size,
                              hipStream_t stream) {
    const float* xyz   = (const float*)d_in[0];
    const float* pts   = (const float*)d_in[1];
    const int*   fps   = (const int*)d_in[2];
    const float* w0    = (const float*)d_in[3];
    const float* b0    = (const float*)d_in[4];
    const float* g0p   = (const float*)d_in[5];
    const float* be0   = (const float*)d_in[6];
    const float* w1    = (const float*)d_in[7];
    const float* b1    = (const float*)d_in[8];
    const float* g1p   = (const float*)d_in[9];
    const float* be1   = (const float*)d_in[10];
    const float* w2    = (const float*)d_in[11];
    const float* b2    = (const float*)d_in[12];
    const float* g2p   = (const float*)d_in[13];
    const float* be2   = (const float*)d_in[14];
    float* out = (float*)d_out;

    char* ws = (char*)d_ws;
    int*   knn   = (int*)(ws);                                   // 1 MB
    float* sum   = (float*)(ws + (1u << 20));                    // 256 f
    float* sumsq = sum + 256;
    float* scale = sum + 512;
    float* shift = sum + 768;
    float* wp0   = sum + 1024;                                   // 64*68 f
    float* bufA  = (float*)(ws + (2u << 20));                    // 128*MTOT f (134 MB)
    float* bufB  = bufA + (size_t)128 * MTOT;                    // 64*MTOT f (67 MB)

    float* newxyz = out;                      // (B, NPOINT, 3)
    float* newpts = out + B_ * NPOINT_ * 3;   // (B, NPOINT, 256)

    newxyz_kernel<<<(B_ * NPOINT_ + 255) / 256, 256, 0, stream>>>(xyz, fps, newxyz);
    padw0_kernel<<<(64 * K0PAD + 255) / 256, 256, 0, stream>>>(w0, wp0);
    knn_kernel<<<B_ * NPOINT_, 256, 0, stream>>>(xyz, fps, knn);
    gather_kernel<<<MTOT / 256, 256, 0, stream>>>(xyz, pts, knn, bufA);

    dim3 grid0(NPOINT_ / 16, 64 / 16, B_);
    dim3 grid1(NPOINT_ / 16, 128 / 16, B_);
    dim3 grid2(NPOINT_ / 16, 256 / 16, B_);

    // Layer 0: x0(bufA,68ch) -> x1(bufB,64ch)
    zero_stats_kernel<<<1, 256, 0, stream>>>(sum, sumsq);
    conv_kernel<K0PAD, 64, 0><<<grid0, 32, 0, stream>>>(bufA, wp0, b0, scale, shift, sum, sumsq, nullptr);
    finalize_kernel<<<1, 256, 0, stream>>>(sum, sumsq, g0p, be0, scale, shift, 64);
    conv_kernel<K0PAD, 64, 1><<<grid0, 32, 0, stream>>>(bufA, wp0, b0, scale, shift, sum, sumsq, bufB);

    // Layer 1: x1(bufB,64ch) -> x2(bufA,128ch)
    zero_stats_kernel<<<1, 256, 0, stream>>>(sum, sumsq);
    conv_kernel<64, 128, 0><<<grid1, 32, 0, stream>>>(bufB, w1, b1, scale, shift, sum, sumsq, nullptr);
    finalize_kernel<<<1, 256, 0, stream>>>(sum, sumsq, g1p, be1, scale, shift, 128);
    conv_kernel<64, 128, 1><<<grid1, 32, 0, stream>>>(bufB, w1, b1, scale, shift, sum, sumsq, bufA);

    // Layer 2: x2(bufA,128ch) -> normalize + relu + max over s -> (B,NPOINT,256)
    zero_stats_kernel<<<1, 256, 0, stream>>>(sum, sumsq);
    conv_kernel<128, 256, 0><<<grid2, 32, 0, stream>>>(bufA, w2, b2, scale, shift, sum, sumsq, nullptr);
    finalize_kernel<<<1, 256, 0, stream>>>(sum, sumsq, g2p, be2, scale, shift, 256);
    conv_kernel<128, 256, 2><<<grid2, 32, 0, stream>>>(bufA, w2, b2, scale, shift, sum, sumsq, newpts);
}